// HybridMatrixNeuralODE_73778948210696
// MI455X (gfx1250) — compile-verified
//
#include <hip/hip_runtime.h>
#include <stdint.h>

typedef __attribute__((ext_vector_type(2))) float v2f;
typedef __attribute__((ext_vector_type(8))) float v8f;

static constexpr int Hdim = 1024;   // H == O == K == N == all leading dims
static constexpr int Bdim = 16384;  // batch rows
static constexpr int WAVES_PER_BLOCK = 8;
static constexpr int THREADS = WAVES_PER_BLOCK * 32;  // wave32

// D = A(16x4,f32) * B(4x16,f32) + C(16x16,f32) on the CDNA5 fp32 matrix pipe.
__device__ __forceinline__ v8f wmma4(v2f a, v2f b, v8f c) {
  return __builtin_amdgcn_wmma_f32_16x16x4_f32(
      /*neg_a=*/false, a, /*neg_b=*/false, b,
      /*c_mod=*/(short)0, c, /*reuse_a=*/false, /*reuse_b=*/false);
}

// Aligned v2f load helper (8B-aligned by construction everywhere below).
__device__ __forceinline__ v2f load2(const float* p) {
  return *(const v2f*)__builtin_assume_aligned(p, 8);
}

// P = I + A  (Hdim x Hdim)
__global__ __launch_bounds__(256) void init_P_kernel(const float* __restrict__ A,
                                                     float* __restrict__ P) {
  int i = blockIdx.x * blockDim.x + threadIdx.x;
  if (i < Hdim * Hdim) {
    int row = i >> 10, col = i & (Hdim - 1);
    P[i] = A[i] + (row == col ? 1.0f : 0.0f);
  }
}

// Pack B (KxN, optionally B = Bsrc^T) into WMMA-B-operand-native layout:
// per k-group g (4 rows) and column n, store [B[4g],B[4g+2],B[4g+1],B[4g+3]]
// contiguously, so a lane's v2f B-frag is one aligned b64 load:
//   frag(g,n,half) = Bs[((g*N + n) << 2) + 2*half .. +1]
template <bool TRANS>
__global__ __launch_bounds__(256) void swizzleB_kernel(const float* __restrict__ Bsrc,
                                                       float* __restrict__ Bs) {
  int i = blockIdx.x * blockDim.x + threadIdx.x;
  if (i >= Hdim * Hdim) return;
  int k = i >> 10, n = i & (Hdim - 1);
  float v = TRANS ? Bsrc[(uint32_t)n * Hdim + (uint32_t)k]
                  : Bsrc[(uint32_t)k * Hdim + (uint32_t)n];
  uint32_t pos = (((uint32_t)(k >> 2) * Hdim + (uint32_t)n) << 2) +
                 (uint32_t)((k & 1) * 2 + ((k >> 1) & 1));  // 0->0,1->2,2->1,3->3
  Bs[pos] = v;
}

// fp32 WMMA GEMM with pre-swizzled B, all dims/lds = Hdim (compile-time):
//   D(M x 1024) = Amat(M x 1024) * B(1024 x 1024) [+ C] [ReLU]
//   CMODE: 0 = zero init, 1 = Cin row-major matrix, 2 = Cin bias row
//   DSWIZ: store D in the swizzled B-operand layout (feeds a later GEMM)
// Each wave computes a 64x64 macro-tile = 4x4 WMMA tiles. Per k-group:
// 16 wmma fed by 8 global_load_b64 whose tile deltas are instruction
// immediates off two running 64-bit pointers (2 pointer bumps / k-group).
template <int CMODE, bool RELU, bool DSWIZ>
__global__ __launch_bounds__(256) void gemm_wmma_kernel(
    const float* __restrict__ Amat,
    const float* __restrict__ Bs,
    const float* __restrict__ Cin,
    float* __restrict__ Dout,
    int Mt4, int Nt4) {
  constexpr uint32_t LD = Hdim;
  const int lane = threadIdx.x & 31;
  const int wave = threadIdx.x >> 5;
  const int id = blockIdx.x * WAVES_PER_BLOCK + wave;
  if (id >= Mt4 * Nt4) return;  // wave-uniform: EXEC stays all-ones below

  const int mm = id / Nt4;
  const int nn = id - mm * Nt4;
  const int m0 = mm * 64;
  const int n0 = nn * 64;
  const int half = lane >> 4;  // 0: lanes 0-15, 1: lanes 16-31
  const int r = lane & 15;

  v8f acc[4][4];
#pragma unroll
  for (int mt = 0; mt < 4; ++mt)
#pragma unroll
    for (int nt = 0; nt < 4; ++nt) acc[mt][nt] = v8f{};

  if (CMODE == 1) {
#pragma unroll
    for (int mt = 0; mt < 4; ++mt)
#pragma unroll
      for (int v = 0; v < 8; ++v) {
        const uint32_t row = (uint32_t)(m0 + mt * 16 + v + 8 * half) * LD;
#pragma unroll
        for (int nt = 0; nt < 4; ++nt)
          acc[mt][nt][v] = Cin[row + (uint32_t)(n0 + nt * 16 + r)];
      }
  } else if (CMODE == 2) {
    float bv[4];
#pragma unroll
    for (int nt = 0; nt < 4; ++nt) bv[nt] = Cin[n0 + nt * 16 + r];
#pragma unroll
    for (int mt = 0; mt < 4; ++mt)
#pragma unroll
      for (int nt = 0; nt < 4; ++nt)
#pragma unroll
        for (int v = 0; v < 8; ++v) acc[mt][nt][v] = bv[nt];
  }

  // Two running pointers; all tile deltas below are compile-time constants
  // (mt*16*LD floats = 64KB steps, nt*64 floats = 256B steps) that fold into
  // the 24-bit global_load immediate offset.
  const float* ap = Amat + (uint32_t)(m0 + r) * LD + (uint32_t)(2 * half);
  const float* bp = Bs + ((uint32_t)(n0 + r) << 2) + (uint32_t)(2 * half);

#pragma unroll 2
  for (int kg = 0; kg < Hdim / 4; ++kg) {
    v2f a[4], b[4];
#pragma unroll
    for (int mt = 0; mt < 4; ++mt) a[mt] = load2(ap + mt * 16 * (int)LD);
#pragma unroll
    for (int nt = 0; nt < 4; ++nt) b[nt] = load2(bp + nt * 64);
#pragma unroll
    for (int mt = 0; mt < 4; ++mt)
#pragma unroll
      for (int nt = 0; nt < 4; ++nt)
        acc[mt][nt] = wmma4(a[mt], b[nt], acc[mt][nt]);

    ap += 4;        // next k-group in A rows (16 bytes)
    bp += 4 * LD;   // next k-group in swizzled B (16 KB)
  }

#pragma unroll
  for (int mt = 0; mt < 4; ++mt)
#pragma unroll
    for (int v = 0; v < 8; ++v) {
      const uint32_t mrow = (uint32_t)(m0 + mt * 16 + v + 8 * half);
#pragma unroll
      for (int nt = 0; nt < 4; ++nt) {
        float e = acc[mt][nt][v];
        if (RELU) e = fmaxf(e, 0.0f);
        const uint32_t ncol = (uint32_t)(n0 + nt * 16 + r);
        if (DSWIZ) {
          // (mrow % 4) == (v % 4) since m0%64==0, mt*16%4==0, 8*half%4==0
          const uint32_t pos = (((mrow >> 2) * LD + ncol) << 2) +
                               (uint32_t)((v & 1) * 2 + ((v >> 1) & 1));
          Dout[pos] = e;
        } else {
          Dout[mrow * LD + ncol] = e;
        }
      }
    }
}

extern "C" void kernel_launch(void* const* d_in, const int* in_sizes, int n_in,
                              void* d_out, int out_size, void* d_ws, size_t ws_size,
                              hipStream_t stream) {
  (void)in_sizes; (void)n_in; (void)out_size; (void)ws_size;
  const float* x    = (const float*)d_in[0];  // (16384, 1024)
  const float* A    = (const float*)d_in[1];  // (1024, 1024)
  const float* W    = (const float*)d_in[2];  // (1024, 1024)
  const float* bias = (const float*)d_in[3];  // (1024,)
  float* out = (float*)d_out;

  const size_t MAT = (size_t)Hdim * Hdim;
  float* P0  = (float*)d_ws;        // 4 MB ping
  float* P1  = P0 + MAT;            // 4 MB pong
  float* Asw = P1 + MAT;            // 4 MB: A swizzled as B-operand
  float* Wsw = Asw + MAT;           // 4 MB: W^T swizzled as B-operand

  const int eBlocks = (Hdim * Hdim + 255) / 256;

  // Phase 0: P = I + A, and pack the two constant B operands.
  init_P_kernel<<<eBlocks, 256, 0, stream>>>(A, P0);
  swizzleB_kernel<false><<<eBlocks, 256, 0, stream>>>(A, Asw);
  swizzleB_kernel<true><<<eBlocks, 256, 0, stream>>>(W, Wsw);

  // Phase 1: nine updates  P <- P + P*A   =>  P = (I+A)^10
  const int sMt4 = Hdim / 64, sNt4 = Hdim / 64;
  const int sBlocks = (sMt4 * sNt4 + WAVES_PER_BLOCK - 1) / WAVES_PER_BLOCK;
  float* src = P0;
  float* dst = P1;
  for (int s = 0; s < 9; ++s) {
    gemm_wmma_kernel<1, false, false><<<sBlocks, THREADS, 0, stream>>>(
        src, Asw, src, dst, sMt4, sNt4);
    float* t = src; src = dst; dst = t;
  }

  // Phase 2: C = P10 * W^T, stored directly in swizzled B-operand layout.
  gemm_wmma_kernel<0, false, true><<<sBlocks, THREADS, 0, stream>>>(
      src, Wsw, nullptr, dst, sMt4, sNt4);
  float* Cswz = dst;

  // Phase 3: out = ReLU(x * C + bias) — single streaming GEMM, 0.5 loads/wmma.
  const int bMt4 = Bdim / 64, bNt4 = Hdim / 64;
  const int bBlocks = (bMt4 * bNt4 + WAVES_PER_BLOCK - 1) / WAVES_PER_BLOCK;
  gemm_wmma_kernel<2, true, false><<<bBlocks, THREADS, 0, stream>>>(
      x, Cswz, bias, out, bMt4, bNt4);
}